// computeMaskedOutput_47382079209764
// MI455X (gfx1250) — compile-verified
//
#include <hip/hip_runtime.h>
#include <hip/hip_bf16.h>
#include <stdint.h>

// Problem constants (from reference): B=64, H=28, W=28, C=512
#define Bn   64
#define Hn   28
#define Wn   28
#define Cn   512
#define HW   784          // Hn*Wn
#define CT   32           // channels per block tile
#define TPB  256          // 8 waves of 32
#define PITCH 788         // LDS pitch (dwords) for template rows: 788*4=3152B (16B aligned),
                          // stride mod 64 banks = 20 -> at most 2-way read conflict

__global__ __launch_bounds__(TPB)
void peak_template_relu_47382079209764_kernel(const float* __restrict__ x,
                                              const float* __restrict__ tp,
                                              float* __restrict__ out)
{
    // 32 template rows staged in LDS: 32*788*4 = 100864 bytes (CDNA5 WGP has 320KB LDS)
    __shared__ __attribute__((aligned(16))) float s_tmpl[CT * PITCH];
    __shared__ int s_idx[CT];
    // Reduction scratch overlaid on s_tmpl: it is dead before the template DMA begins.
    float* s_redv = s_tmpl;                  // 32*32 floats (4KB)
    int*   s_redi = (int*)(s_tmpl + 1024);   // 32*32 ints   (4KB)

    const int t  = threadIdx.x;
    const int c0 = (t & 7) * 4;        // this lane owns 4 consecutive channels
    const int rg = t >> 3;             // position group in [0,32); 4 distinct values per wave
    const int bk = blockIdx.x;
    const int b  = bk >> 4;            // 16 channel tiles per batch
    const int cb = (bk & 15) * CT;     // tile channel base

    // base points at (b, p=0, cb+c0); advance by p*Cn
    const size_t base = (size_t)b * HW * Cn + (size_t)(cb + c0);

    // ---- Pass 1: per-(b,c) argmax over 784 positions, 4 channels/lane, b128 loads ----
    float best[4];
    int   bidx[4];
    #pragma unroll
    for (int j = 0; j < 4; ++j) { best[j] = -3.402823466e+38f; bidx[j] = 0; }
    for (int p = rg; p < HW; p += 32) {
        float4 v = *reinterpret_cast<const float4*>(x + base + (size_t)p * Cn);
        float vv[4] = {v.x, v.y, v.z, v.w};
        #pragma unroll
        for (int j = 0; j < 4; ++j)
            if (vv[j] > best[j]) { best[j] = vv[j]; bidx[j] = p; }  // '>' keeps first occurrence
    }
    #pragma unroll
    for (int j = 0; j < 4; ++j) {
        s_redv[rg * CT + c0 + j] = best[j];
        s_redi[rg * CT + c0 + j] = bidx[j];
    }
    __syncthreads();
    if (t < CT) {                       // one lane per channel reduces 32 groups
        float bv = s_redv[t];
        int   bi = s_redi[t];
        for (int g = 1; g < 32; ++g) {
            float v = s_redv[g * CT + t];
            int   i = s_redi[g * CT + t];
            if (v > bv || (v == bv && i < bi)) { bv = v; bi = i; }
        }
        s_idx[t] = bi;                  // flat spatial index of the peak
    }
    __syncthreads();                    // reduction reads done; s_tmpl space reusable

    // ---- Pass 2a: async DMA of the 32 gathered template rows (784 floats) into LDS ----
    // Each b128 op moves 16B/lane; row bases are 16B aligned (784*4 = 196*16).
    const int chunksPerRow = HW / 4;                 // 196
    const int totalChunks  = CT * chunksPerRow;      // 6272 = 24.5 * 256 (wave-uniform tail)
    for (int q = t; q < totalChunks; q += TPB) {
        int cc = q / chunksPerRow;
        int k  = q - cc * chunksPerRow;
        const float* src = tp + ((size_t)b * HW + s_idx[cc]) * (size_t)HW + (size_t)k * 4;
        uint32_t lds_off = (uint32_t)(uintptr_t)(&s_tmpl[cc * PITCH + k * 4]);
        unsigned long long ga = (unsigned long long)(uintptr_t)src;
        asm volatile("global_load_async_to_lds_b128 %0, %1, off"
                     :: "v"(lds_off), "v"(ga)
                     : "memory");
    }
    asm volatile("s_wait_asynccnt 0" ::: "memory");  // drain this wave's ASYNCcnt
    __syncthreads();                                  // all waves' LDS writes visible

    // ---- Pass 2b: out = relu(x * template), b128 in / b128 out ----
    for (int p = rg; p < HW; p += 32) {
        float4 v = *reinterpret_cast<const float4*>(x + base + (size_t)p * Cn); // L2-hot
        float t0 = s_tmpl[(c0 + 0) * PITCH + p];
        float t1 = s_tmpl[(c0 + 1) * PITCH + p];
        float t2 = s_tmpl[(c0 + 2) * PITCH + p];
        float t3 = s_tmpl[(c0 + 3) * PITCH + p];
        float4 o;
        o.x = fmaxf(v.x * t0, 0.0f);
        o.y = fmaxf(v.y * t1, 0.0f);
        o.z = fmaxf(v.z * t2, 0.0f);
        o.w = fmaxf(v.w * t3, 0.0f);
        *reinterpret_cast<float4*>(out + base + (size_t)p * Cn) = o;
    }
}

extern "C" void kernel_launch(void* const* d_in, const int* in_sizes, int n_in,
                              void* d_out, int out_size, void* d_ws, size_t ws_size,
                              hipStream_t stream) {
    (void)in_sizes; (void)n_in; (void)d_ws; (void)ws_size; (void)out_size;
    const float* x  = (const float*)d_in[0];   // [B,H,W,C] f32
    const float* tp = (const float*)d_in[1];   // [B,H,W,H,W] f32
    float* out = (float*)d_out;                // [B,H,W,C] f32
    dim3 grid(Bn * (Cn / CT));                 // 64 batches * 16 channel tiles = 1024
    peak_template_relu_47382079209764_kernel<<<grid, TPB, 0, stream>>>(x, tp, out);
}